// MPMModel_58213986730458
// MI455X (gfx1250) — compile-verified
//
#include <hip/hip_runtime.h>

// ---------------- MPM constants (match reference) ----------------
#define NG 512
#define NCELL (NG * NG)                 // 262144
#define CPB 32                          // base cells owned per P2G block
#define NBLK_P2G (NCELL / CPB)          // 8192
#define TILE (CPB + 2 * NG + 2)         // 1058 linear cells covered by a block's stencils
#define TDM_CAP 2048                    // staged order[] indices per block (8 KB LDS)
static constexpr float DXc     = 1.0f / 512.0f;
static constexpr float INV_DXc = 512.0f;
static constexpr float DTc     = 1e-4f;
static constexpr float P_VOLc  = (DXc * 0.5f) * (DXc * 0.5f);
static constexpr float P_MASSc = P_VOLc * 1.0f;
static constexpr float GRAVc   = 10.0f;

typedef __attribute__((ext_vector_type(4))) unsigned v4u;
typedef __attribute__((ext_vector_type(8))) int      v8i;
typedef __attribute__((ext_vector_type(4))) int      v4i;

__device__ __forceinline__ void quad_w(float fx, float w[3]) {
    w[0] = 0.5f * (1.5f - fx) * (1.5f - fx);
    w[1] = 0.75f - (fx - 1.0f) * (fx - 1.0f);
    w[2] = 0.5f * (fx - 0.5f) * (fx - 0.5f);
}

__device__ __forceinline__ void atomic_add_dev_f32(float* p, float v) {
    (void)__hip_atomic_fetch_add(p, v, __ATOMIC_RELAXED, __HIP_MEMORY_SCOPE_AGENT);
}
__device__ __forceinline__ void atomic_add_lds_f32(float* p, float v) {
    (void)__hip_atomic_fetch_add(p, v, __ATOMIC_RELAXED, __HIP_MEMORY_SCOPE_WORKGROUP);
}

// ---- shared per-particle physics: F-update, polar, stress, affine ----
struct P2GVals {
    int bx, by;
    float fx0, fx1;
    float wx[3], wy[3];
    float af00, af01, af10, af11, mvx, mvy;
    float F00, F01, F10, F11;
    float jp;
};

__device__ __forceinline__ P2GVals p2g_core(
    int p, const float2* __restrict__ x, const float2* __restrict__ v,
    const float4* __restrict__ C, const float4* __restrict__ F,
    const float* __restrict__ Jp, float mu0, float la0)
{
    P2GVals o;
    const float2 xp = x[p];
    const float2 vp = v[p];
    const float4 Cp = C[p];   // row-major 2x2
    const float4 Fp = F[p];
    o.jp = Jp[p];

    float fx0 = xp.x * INV_DXc;
    float fx1 = xp.y * INV_DXc;
    o.bx = (int)(fx0 - 0.5f);           // positive domain -> trunc == floor
    o.by = (int)(fx1 - 0.5f);
    o.fx0 = fx0 - (float)o.bx;
    o.fx1 = fx1 - (float)o.by;
    quad_w(o.fx0, o.wx);
    quad_w(o.fx1, o.wy);

    // F <- F + dt * C @ F
    o.F00 = Fp.x + DTc * (Cp.x * Fp.x + Cp.y * Fp.z);
    o.F01 = Fp.y + DTc * (Cp.x * Fp.y + Cp.y * Fp.w);
    o.F10 = Fp.z + DTc * (Cp.z * Fp.x + Cp.w * Fp.z);
    o.F11 = Fp.w + DTc * (Cp.z * Fp.y + Cp.w * Fp.w);

    const float h  = expf(10.0f * (1.0f - o.jp));
    const float mu = mu0 * h;
    const float la = la0 * h;

    // closed-form 2x2 polar rotation: R = [[c,-s],[s,c]]
    const float a  = o.F00 + o.F11;
    const float b  = o.F10 - o.F01;
    const float r  = sqrtf(a * a + b * b);
    const float cc = a / r;
    const float ss = b / r;
    const float J  = fabsf(o.F00 * o.F11 - o.F01 * o.F10);

    // PF = (F - R) @ F^T
    const float A00 = o.F00 - cc, A01 = o.F01 + ss;
    const float A10 = o.F10 - ss, A11 = o.F11 - cc;
    const float PF00 = A00 * o.F00 + A01 * o.F01;
    const float PF01 = A00 * o.F10 + A01 * o.F11;
    const float PF10 = A10 * o.F00 + A11 * o.F01;
    const float PF11 = A10 * o.F10 + A11 * o.F11;

    const float ld = la * J * (J - 1.0f);
    const float SK = -DTc * P_VOLc * 4.0f * INV_DXc * INV_DXc;
    o.af00 = SK * (2.0f * mu * PF00 + ld) + P_MASSc * Cp.x;
    o.af01 = SK * (2.0f * mu * PF01)      + P_MASSc * Cp.y;
    o.af10 = SK * (2.0f * mu * PF10)      + P_MASSc * Cp.z;
    o.af11 = SK * (2.0f * mu * PF11 + ld) + P_MASSc * Cp.w;
    o.mvx = P_MASSc * vp.x;
    o.mvy = P_MASSc * vp.y;
    return o;
}

// ================= sorted path: LDS-tiled P2G with TDM index staging =================
__global__ void __launch_bounds__(256) k_p2g_sorted(
    const float2* __restrict__ x, const float2* __restrict__ v,
    const float4* __restrict__ C, const float4* __restrict__ F,
    const int* __restrict__ material, const float* __restrict__ Jp,
    const float* __restrict__ Ep, const float* __restrict__ nup,
    const unsigned* __restrict__ order, const unsigned* __restrict__ offsets,
    float* __restrict__ grid, float4* __restrict__ outF,
    int* __restrict__ outMat, float* __restrict__ outJp, int n)
{
    __shared__ float tvx[TILE];
    __shared__ float tvy[TILE];
    __shared__ float tm[TILE];
    __shared__ unsigned sOrder[TDM_CAP];

    const int cell0 = blockIdx.x * CPB;
    const int pbeg  = (int)offsets[cell0];
    const int pend  = (cell0 + CPB < NCELL) ? (int)offsets[cell0 + CPB] : n;
    const int cnt   = pend - pbeg;
    const int staged = cnt < TDM_CAP ? cnt : TDM_CAP;

#if __has_builtin(__builtin_amdgcn_tensor_load_to_lds) && __has_builtin(__builtin_amdgcn_s_wait_tensorcnt)
    // Stage this block's contiguous order[] slice into LDS with the Tensor Data
    // Mover (1-D tile, 4-byte elements), overlapping with tile zeroing below.
    if (threadIdx.x == 0 && staged > 0) {
        const unsigned lds_off = (unsigned)(unsigned long long)(&sOrder[0]);
        const unsigned long long ga = (unsigned long long)(const void*)(order + pbeg);
        const unsigned us = (unsigned)staged;
        v4u g0 = { 1u,                                   // count=1 valid, user D#
                   lds_off,                               // lds_addr
                   (unsigned)(ga & 0xFFFFFFFFu),          // global_addr[31:0]
                   (unsigned)((ga >> 32) & 0x01FFFFFFu) | (2u << 30) }; // [56:32] | type=2
        v8i g1 = { (int)0x00020000u,                      // data_size=4B; no multicast/pad
                   (int)((us & 0xFFFFu) << 16),           // tensor_dim0[15:0]
                   (int)(((us >> 16) & 0xFFFFu) | (1u << 16)), // tensor_dim0[31:16], tensor_dim1=1
                   (int)((us & 0xFFFFu) << 16),           // tile_dim0 = staged
                   0,                                     // tile_dim1=0 (1-D), tile_dim2=0
                   (int)us,                               // tensor_dim0_stride[31:0]
                   0, 0 };
        v4i gz  = { 0, 0, 0, 0 };
        v8i gz8 = { 0, 0, 0, 0, 0, 0, 0, 0 };
        // amdgpu-toolchain (clang-23) arity: 6 args (extra int32x8 group), cpol last
        __builtin_amdgcn_tensor_load_to_lds(g0, g1, gz, gz, gz8, 0);
    }
    for (int t = threadIdx.x; t < TILE; t += blockDim.x) {
        tvx[t] = 0.0f; tvy[t] = 0.0f; tm[t] = 0.0f;
    }
    if (threadIdx.x == 0 && staged > 0) {
        __builtin_amdgcn_s_wait_tensorcnt(0);             // TDM done before barrier
    }
#else
    for (int t = threadIdx.x; t < staged; t += blockDim.x) sOrder[t] = order[pbeg + t];
    for (int t = threadIdx.x; t < TILE; t += blockDim.x) {
        tvx[t] = 0.0f; tvy[t] = 0.0f; tm[t] = 0.0f;
    }
#endif
    __syncthreads();

    const float E  = Ep[0];
    const float nu = nup[0];
    const float mu0 = E / (2.0f * (1.0f + nu));
    const float la0 = E * nu / ((1.0f + nu) * (1.0f - 2.0f * nu));

    for (int s = pbeg + (int)threadIdx.x; s < pend; s += (int)blockDim.x) {
        const int rel = s - pbeg;
        const int p = (int)((rel < staged) ? sOrder[rel] : order[s]);
        const P2GVals o = p2g_core(p, x, v, C, F, Jp, mu0, la0);
        const int local = o.bx * NG + o.by - cell0;   // in [0, CPB)
#pragma unroll
        for (int i = 0; i < 3; ++i) {
            const float dposx = ((float)i - o.fx0) * DXc;
            const float cxi = o.af00 * dposx + o.mvx;
            const float cyi = o.af10 * dposx + o.mvy;
            const int rowoff = local + i * NG;
#pragma unroll
            for (int j = 0; j < 3; ++j) {
                const float dposy = ((float)j - o.fx1) * DXc;
                const float wt = o.wx[i] * o.wy[j];
                const int t = rowoff + j;             // < TILE by construction
                atomic_add_lds_f32(&tvx[t], wt * (cxi + o.af01 * dposy));
                atomic_add_lds_f32(&tvy[t], wt * (cyi + o.af11 * dposy));
                atomic_add_lds_f32(&tm[t],  wt * P_MASSc);
            }
        }
        outF[p]   = make_float4(o.F00, o.F01, o.F10, o.F11);
        outMat[p] = material[p];
        outJp[p]  = o.jp;
    }
    __syncthreads();

    // flush tile -> global grid (halos overlap the next blocks' ranges -> atomics)
    for (int t = threadIdx.x; t < TILE; t += blockDim.x) {
        const float m = tm[t];
        if (m != 0.0f) {
            const int cid = cell0 + t;
            if (cid < NCELL) {
                float* cell = grid + 4 * (size_t)cid;
                atomic_add_dev_f32(cell + 0, tvx[t]);
                atomic_add_dev_f32(cell + 1, tvy[t]);
                atomic_add_dev_f32(cell + 2, m);
            }
        }
    }
}

// ---------------- counting sort: histogram ----------------
__global__ void __launch_bounds__(256) k_hist(
    const float2* __restrict__ x, unsigned* __restrict__ hist, int n)
{
    int p = blockIdx.x * blockDim.x + threadIdx.x;
    if (p >= n) return;
    __builtin_prefetch(&x[p + 4096], 0, 1);
    const float2 xp = x[p];
    const int bx = (int)(xp.x * INV_DXc - 0.5f);
    const int by = (int)(xp.y * INV_DXc - 0.5f);
    (void)__hip_atomic_fetch_add(&hist[bx * NG + by], 1u,
                                 __ATOMIC_RELAXED, __HIP_MEMORY_SCOPE_AGENT);
}

// ---------------- counting sort: 2-level exclusive scan over NCELL ----------------
__global__ void __launch_bounds__(512) k_scan1(
    const unsigned* __restrict__ in, unsigned* __restrict__ out,
    unsigned* __restrict__ partials)
{
    __shared__ unsigned tmp[512];
    const int t = threadIdx.x;
    const int g = blockIdx.x * 512 + t;
    const unsigned vIn = in[g];
    tmp[t] = vIn;
    __syncthreads();
#pragma unroll
    for (int off = 1; off < 512; off <<= 1) {
        const unsigned add = (t >= off) ? tmp[t - off] : 0u;
        __syncthreads();
        tmp[t] += add;
        __syncthreads();
    }
    out[g] = tmp[t] - vIn;                    // exclusive
    if (t == 511) partials[blockIdx.x] = tmp[t];
}

__global__ void __launch_bounds__(512) k_scan2(unsigned* __restrict__ partials)
{
    __shared__ unsigned tmp[512];
    const int t = threadIdx.x;
    const unsigned vIn = partials[t];
    tmp[t] = vIn;
    __syncthreads();
#pragma unroll
    for (int off = 1; off < 512; off <<= 1) {
        const unsigned add = (t >= off) ? tmp[t - off] : 0u;
        __syncthreads();
        tmp[t] += add;
        __syncthreads();
    }
    partials[t] = tmp[t] - vIn;               // exclusive block offsets
}

__global__ void __launch_bounds__(512) k_scan3(
    unsigned* __restrict__ out, unsigned* __restrict__ head,
    const unsigned* __restrict__ partials)
{
    const int g = blockIdx.x * 512 + threadIdx.x;
    const unsigned val = out[g] + partials[blockIdx.x];
    out[g]  = val;                            // preserved range table
    head[g] = val;                            // working copy for scatter
}

// ---------------- counting sort: index scatter ----------------
__global__ void __launch_bounds__(256) k_scatter(
    const float2* __restrict__ x, unsigned* __restrict__ head,
    unsigned* __restrict__ order, int n)
{
    int p = blockIdx.x * blockDim.x + threadIdx.x;
    if (p >= n) return;
    __builtin_prefetch(&x[p + 4096], 0, 1);
    const float2 xp = x[p];
    const int bx = (int)(xp.x * INV_DXc - 0.5f);
    const int by = (int)(xp.y * INV_DXc - 0.5f);
    const unsigned pos = __hip_atomic_fetch_add(&head[bx * NG + by], 1u,
                                                __ATOMIC_RELAXED, __HIP_MEMORY_SCOPE_AGENT);
    order[pos] = (unsigned)p;
}

// ================= fallback: direct-atomic P2G =================
__global__ void __launch_bounds__(256) k_p2g_direct(
    const float2* __restrict__ x, const float2* __restrict__ v,
    const float4* __restrict__ C, const float4* __restrict__ F,
    const int* __restrict__ material, const float* __restrict__ Jp,
    const float* __restrict__ Ep, const float* __restrict__ nup,
    float* __restrict__ grid, float4* __restrict__ outF,
    int* __restrict__ outMat, float* __restrict__ outJp, int n)
{
    int p = blockIdx.x * blockDim.x + threadIdx.x;
    if (p >= n) return;
    __builtin_prefetch(&C[p + 2048], 0, 1);
    __builtin_prefetch(&F[p + 2048], 0, 1);

    const float E  = Ep[0];
    const float nu = nup[0];
    const float mu0 = E / (2.0f * (1.0f + nu));
    const float la0 = E * nu / ((1.0f + nu) * (1.0f - 2.0f * nu));

    const P2GVals o = p2g_core(p, x, v, C, F, Jp, mu0, la0);

#pragma unroll
    for (int i = 0; i < 3; ++i) {
        const float dposx = ((float)i - o.fx0) * DXc;
        const float cxi = o.af00 * dposx + o.mvx;
        const float cyi = o.af10 * dposx + o.mvy;
        float* rowbase = grid + 4 * (size_t)((o.bx + i) * NG + o.by);
#pragma unroll
        for (int j = 0; j < 3; ++j) {
            const float dposy = ((float)j - o.fx1) * DXc;
            const float wt = o.wx[i] * o.wy[j];
            float* cell = rowbase + 4 * j;
            atomic_add_dev_f32(cell + 0, wt * (cxi + o.af01 * dposy));
            atomic_add_dev_f32(cell + 1, wt * (cyi + o.af11 * dposy));
            atomic_add_dev_f32(cell + 2, wt * P_MASSc);
        }
    }
    outF[p]   = make_float4(o.F00, o.F01, o.F10, o.F11);
    outMat[p] = material[p];
    outJp[p]  = o.jp;
}

// ---------------- zero kernels ----------------
__global__ void k_zero_grid(float4* __restrict__ grid) {
    int i = blockIdx.x * blockDim.x + threadIdx.x;
    if (i < NCELL) grid[i] = make_float4(0.0f, 0.0f, 0.0f, 0.0f);
}
__global__ void k_zero_u32(unsigned* __restrict__ a) {
    int i = blockIdx.x * blockDim.x + threadIdx.x;
    if (i < NCELL) a[i] = 0u;
}

// ---------------- grid normalize + gravity + BCs (2-D launch) ----------------
__global__ void k_grid_update(float4* __restrict__ grid) {
    const int gy = blockIdx.x * blockDim.x + threadIdx.x;  // column
    const int gx = blockIdx.y;                             // row
    if (gy >= NG) return;
    const int idx = gx * NG + gy;
    float4 c = grid[idx];
    float vx = c.x, vy = c.y;
    const float m = c.z;
    if (m > 0.0f) {
        const float inv = 1.0f / m;
        vx *= inv;
        vy *= inv;
    }
    vy -= DTc * GRAVc;
    if (gx < 3)       vx = fmaxf(vx, 0.0f);
    if (gx >= NG - 2) vx = fminf(vx, 0.0f);
    if (gy < 3)       vy = fmaxf(vy, 0.0f);
    if (gy >= NG - 2) vy = fminf(vy, 0.0f);
    grid[idx] = make_float4(vx, vy, m, 0.0f);
}

// ---------------- G2P gather, advect ----------------
__global__ void __launch_bounds__(256) k_g2p(
    const float2* __restrict__ x, const float* __restrict__ grid,
    float2* __restrict__ outX, float2* __restrict__ outV,
    float4* __restrict__ outC, int n)
{
    int p = blockIdx.x * blockDim.x + threadIdx.x;
    if (p >= n) return;
    __builtin_prefetch(&x[p + 4096], 0, 1);

    const float2 xp = x[p];
    float fx0 = xp.x * INV_DXc;
    float fx1 = xp.y * INV_DXc;
    const int bx = (int)(fx0 - 0.5f);
    const int by = (int)(fx1 - 0.5f);
    fx0 -= (float)bx;
    fx1 -= (float)by;
    float wx[3], wy[3];
    quad_w(fx0, wx);
    quad_w(fx1, wy);

    const float K = 4.0f * INV_DXc * INV_DXc;
    float nvx = 0.0f, nvy = 0.0f;
    float c00 = 0.0f, c01 = 0.0f, c10 = 0.0f, c11 = 0.0f;

#pragma unroll
    for (int i = 0; i < 3; ++i) {
        const float dposx = ((float)i - fx0) * DXc;
        const float* rowbase = grid + 4 * (size_t)((bx + i) * NG + by);
#pragma unroll
        for (int j = 0; j < 3; ++j) {
            const float dposy = ((float)j - fx1) * DXc;
            const float wt = wx[i] * wy[j];
            const float2 gv = *(const float2*)(rowbase + 4 * j);
            nvx += wt * gv.x;
            nvy += wt * gv.y;
            const float kwx = K * wt * gv.x;
            const float kwy = K * wt * gv.y;
            c00 += kwx * dposx;
            c01 += kwx * dposy;
            c10 += kwy * dposx;
            c11 += kwy * dposy;
        }
    }
    outX[p] = make_float2(xp.x + DTc * nvx, xp.y + DTc * nvy);
    outV[p] = make_float2(nvx, nvy);
    outC[p] = make_float4(c00, c01, c10, c11);
}

// ---------------- launch ----------------
extern "C" void kernel_launch(void* const* d_in, const int* in_sizes, int n_in,
                              void* d_out, int out_size, void* d_ws, size_t ws_size,
                              hipStream_t stream) {
    const float2* x   = (const float2*)d_in[0];
    const float2* v   = (const float2*)d_in[1];
    const float4* C   = (const float4*)d_in[2];
    const float4* F   = (const float4*)d_in[3];
    const int*    mat = (const int*)d_in[4];
    const float*  Jp  = (const float*)d_in[5];
    const float*  E   = (const float*)d_in[6];
    const float*  nu  = (const float*)d_in[7];

    const int n = in_sizes[0] / 2;            // N_PART
    float* out  = (float*)d_out;
    float*  outX = out;                               // n*2
    float*  outV = out + (size_t)2 * n;               // n*2
    float4* outC = (float4*)(out + (size_t)4 * n);    // n*4
    float4* outF = (float4*)(out + (size_t)8 * n);    // n*4
    int*    outM = (int*)  (out + (size_t)12 * n);    // n (raw int bits)
    float*  outJ = out + (size_t)13 * n;              // n

    // workspace layout
    float*    grid     = (float*)d_ws;                // 4*NCELL floats (4 MB, L2-resident)
    unsigned* hist     = (unsigned*)(grid + 4 * (size_t)NCELL);  // NCELL
    unsigned* offsets  = hist + NCELL;                // NCELL
    unsigned* head     = offsets + NCELL;             // NCELL
    unsigned* partials = head + NCELL;                // 512
    unsigned* order    = partials + 512;              // n
    const size_t ws_need = (4 * (size_t)NCELL + 3 * (size_t)NCELL + 512 + (size_t)n)
                           * sizeof(unsigned);

    const int BS = 256;
    const int pblk = (n + BS - 1) / BS;
    const int cblk = (NCELL + BS - 1) / BS;

    k_zero_grid<<<cblk, BS, 0, stream>>>((float4*)grid);

    if (ws_size >= ws_need) {
        // counting sort by base cell, then LDS-tiled P2G (27x fewer global atomics)
        k_zero_u32<<<cblk, BS, 0, stream>>>(hist);
        k_hist<<<pblk, BS, 0, stream>>>(x, hist, n);
        k_scan1<<<NCELL / 512, 512, 0, stream>>>(hist, offsets, partials);
        k_scan2<<<1, 512, 0, stream>>>(partials);
        k_scan3<<<NCELL / 512, 512, 0, stream>>>(offsets, head, partials);
        k_scatter<<<pblk, BS, 0, stream>>>(x, head, order, n);
        k_p2g_sorted<<<NBLK_P2G, BS, 0, stream>>>(x, v, C, F, mat, Jp, E, nu,
                                                  order, offsets, grid,
                                                  outF, outM, outJ, n);
    } else {
        k_p2g_direct<<<pblk, BS, 0, stream>>>(x, v, C, F, mat, Jp, E, nu,
                                              grid, outF, outM, outJ, n);
    }

    k_grid_update<<<dim3((NG + BS - 1) / BS, NG), BS, 0, stream>>>((float4*)grid);
    k_g2p<<<pblk, BS, 0, stream>>>(x, grid, (float2*)outX, (float2*)outV, outC, n);
}